// DenseGrid_438086664220
// MI455X (gfx1250) — compile-verified
//
#include <hip/hip_runtime.h>

// Multi-LOD dense-grid trilinear interpolation for MI455X (gfx1250).
// Memory-bound gather kernel: no matmul structure (per-point private corner
// matrices), so no WMMA. Optimization levers: b128 gathers, CDNA5 temporal
// hints (NT for the >L2 grid4 + streams, RT for L2-resident grids), LDS
// residency for the 64KB LOD0 grid filled via async global->LDS DMA, and
// 2 points/thread to amortize the LDS fill and double per-thread MLP.

typedef float v4f __attribute__((ext_vector_type(4)));
typedef int   v4i __attribute__((ext_vector_type(4)));

#define AS1 __attribute__((address_space(1)))
#define AS3 __attribute__((address_space(3)))

#if defined(__gfx1250__) &&                                         \
    __has_builtin(__builtin_amdgcn_global_load_async_to_lds_b128) && \
    __has_builtin(__builtin_amdgcn_s_wait_asynccnt)
#define USE_ASYNC_LDS 1
#else
#define USE_ASYNC_LDS 0
#endif

static __device__ __forceinline__ v4f ntload4(const v4f* p) {
  return __builtin_nontemporal_load(p);
}

template <int RES, bool NT>
static __device__ __forceinline__ v4f trilerp(const v4f* __restrict__ g,
                                              float px, float py, float pz) {
  constexpr float RM1 = (float)(RES - 1);
  // match reference: c = clip(p*(res-1), 0, res-1); i0 = floor(c); f = c - i0
  float cx = fminf(fmaxf(px * RM1, 0.0f), RM1);
  float cy = fminf(fmaxf(py * RM1, 0.0f), RM1);
  float cz = fminf(fmaxf(pz * RM1, 0.0f), RM1);
  float x0f = floorf(cx), y0f = floorf(cy), z0f = floorf(cz);
  float fx = cx - x0f, fy = cy - y0f, fz = cz - z0f;
  int x0 = (int)x0f, y0 = (int)y0f, z0 = (int)z0f;
  int x1 = min(x0 + 1, RES - 1);
  int y1 = min(y0 + 1, RES - 1);
  int z1 = min(z0 + 1, RES - 1);

  int b00 = (x0 * RES + y0) * RES;
  int b01 = (x0 * RES + y1) * RES;
  int b10 = (x1 * RES + y0) * RES;
  int b11 = (x1 * RES + y1) * RES;

  v4f c000, c001, c010, c011, c100, c101, c110, c111;
  if constexpr (NT) {   // grid larger than L2: don't pollute the hot LODs
    c000 = ntload4(g + (b00 + z0)); c001 = ntload4(g + (b00 + z1));
    c010 = ntload4(g + (b01 + z0)); c011 = ntload4(g + (b01 + z1));
    c100 = ntload4(g + (b10 + z0)); c101 = ntload4(g + (b10 + z1));
    c110 = ntload4(g + (b11 + z0)); c111 = ntload4(g + (b11 + z1));
  } else {
    c000 = g[b00 + z0]; c001 = g[b00 + z1];
    c010 = g[b01 + z0]; c011 = g[b01 + z1];
    c100 = g[b10 + z0]; c101 = g[b10 + z1];
    c110 = g[b11 + z0]; c111 = g[b11 + z1];
  }

  float gx = 1.0f - fx, gy = 1.0f - fy, gz = 1.0f - fz;
  float gxgy = gx * gy, gxfy = gx * fy, fxgy = fx * gy, fxfy = fx * fy;

  // same corner/summation order as the reference expression
  v4f acc = c000 * (gxgy * gz);
  acc += c001 * (gxgy * fz);
  acc += c010 * (gxfy * gz);
  acc += c011 * (gxfy * fz);
  acc += c100 * (fxgy * gz);
  acc += c101 * (fxgy * fz);
  acc += c110 * (fxfy * gz);
  acc += c111 * (fxfy * fz);
  return acc;
}

static __device__ __forceinline__ void
process_point(int pid, int n,
              const float* __restrict__ pts,
              const v4f* __restrict__ sh_g0,
              const float* __restrict__ g1, const float* __restrict__ g2,
              const float* __restrict__ g3, const float* __restrict__ g4,
              float* __restrict__ out) {
  if (pid >= n) return;

  // points are streamed once -> NT loads
  float px = __builtin_nontemporal_load(pts + 3 * (size_t)pid + 0);
  float py = __builtin_nontemporal_load(pts + 3 * (size_t)pid + 1);
  float pz = __builtin_nontemporal_load(pts + 3 * (size_t)pid + 2);

  v4f f0 = trilerp<16,  false>(sh_g0,          px, py, pz);  // LDS
  v4f f1 = trilerp<32,  false>((const v4f*)g1, px, py, pz);  // 512 KB, L2-hot
  v4f f2 = trilerp<64,  false>((const v4f*)g2, px, py, pz);  // 4 MB, L2-hot
  v4f f3 = trilerp<128, false>((const v4f*)g3, px, py, pz);  // 32 MB, L2-hot
  v4f f4 = trilerp<256, true >((const v4f*)g4, px, py, pz);  // 256 MB > L2: NT

  // row = 20 floats = 80 B (16-B aligned for every pid); write-once stream -> NT
  v4f* row = (v4f*)(out + (size_t)pid * 20);
  __builtin_nontemporal_store(f0, row + 0);
  __builtin_nontemporal_store(f1, row + 1);
  __builtin_nontemporal_store(f2, row + 2);
  __builtin_nontemporal_store(f3, row + 3);
  __builtin_nontemporal_store(f4, row + 4);
}

__global__ __launch_bounds__(256) void
dense_grid_mlod_kernel(const float* __restrict__ pts,
                       const float* __restrict__ g0,
                       const float* __restrict__ g1,
                       const float* __restrict__ g2,
                       const float* __restrict__ g3,
                       const float* __restrict__ g4,
                       float* __restrict__ out, int n) {
  // LOD0 grid is exactly 16^3 * 4 floats = 64 KB -> keep it in LDS
  // (320 KB per WGP on CDNA5, so this still allows multi-WG residency).
  __shared__ v4f sh_g0[16 * 16 * 16];

  {
    const v4f* g0v = (const v4f*)g0;
#if USE_ASYNC_LDS
    // CDNA5 direct global->LDS DMA path: no VGPR round-trip, single
    // ASYNCcnt wait instead of per-iteration loadcnt waits.
    #pragma unroll
    for (int k = 0; k < 16; ++k) {
      int i = (int)threadIdx.x + k * 256;
      AS1 v4i* gsrc = (AS1 v4i*)const_cast<v4f*>(g0v + i);
      AS3 v4i* ldst = (AS3 v4i*)(sh_g0 + i);
      __builtin_amdgcn_global_load_async_to_lds_b128(gsrc, ldst,
                                                     /*offset=*/0, /*cpol=*/0);
    }
    __builtin_amdgcn_s_wait_asynccnt(0);
#else
    #pragma unroll
    for (int k = 0; k < 16; ++k) {
      int i = (int)threadIdx.x + k * 256;
      sh_g0[i] = g0v[i];
    }
#endif
  }
  __syncthreads();

  // Two points per thread: halves the per-point LDS-fill overhead and
  // doubles per-thread memory-level parallelism.
  int base = blockIdx.x * 512 + (int)threadIdx.x;
  process_point(base,       n, pts, sh_g0, g1, g2, g3, g4, out);
  process_point(base + 256, n, pts, sh_g0, g1, g2, g3, g4, out);
}

extern "C" void kernel_launch(void* const* d_in, const int* in_sizes, int n_in,
                              void* d_out, int out_size, void* d_ws, size_t ws_size,
                              hipStream_t stream) {
  (void)n_in; (void)out_size; (void)d_ws; (void)ws_size;
  const float* pts = (const float*)d_in[0];
  const float* g0  = (const float*)d_in[1];
  const float* g1  = (const float*)d_in[2];
  const float* g2  = (const float*)d_in[3];
  const float* g3  = (const float*)d_in[4];
  const float* g4  = (const float*)d_in[5];
  float* out = (float*)d_out;

  int n = in_sizes[0] / 3;
  int blocks = (n + 511) / 512;
  hipLaunchKernelGGL(dense_grid_mlod_kernel, dim3(blocks), dim3(256), 0, stream,
                     pts, g0, g1, g2, g3, g4, out, n);
}